// ROIPooling_90486370992180
// MI455X (gfx1250) — compile-verified
//
#include <hip/hip_runtime.h>
#include <stdint.h>

// ROI max-pool for MI455X (gfx1250, wave32).
//
// Reference shapes:
//   rois     : [B=2, N=64, 4]  f32   (x1,y1,x2,y2 in image pixels, stride 16)
//   conv_out : [B=2, C=256, H=64, W=64] f32
//   out      : [B, N, C, 7, 7] f32
//
// Strategy: one 256-thread workgroup per (b, n, 4-channel chunk).
//  1) all threads compute ROI window metadata (trunc coords, 7-step expand,
//     kernel=ceil(L/7), stride=floor(L/7)) -- ~30 VALU ops, redundant per thread.
//  2) stage rows [y1..y2] (full 64-float width, 256B contiguous per row) of the
//     4 channels into LDS with GLOBAL_LOAD_ASYNC_TO_LDS_B128 (CDNA5 async path,
//     tracked by ASYNCcnt; 16B per lane per instruction, fully coalesced).
//  3) s_wait_asynccnt 0 + barrier, then 4*49=196 threads each reduce a
//     kernel_h x kernel_w window from LDS with fmaxf and store one output.

#define ANCHOR_STRIDE 16
#define POOL 7
#define B_ 2
#define N_ 64
#define C_ 256
#define H_ 64
#define W_ 64
#define CPB 4                 // channels per block; LDS = 4*64*64*4B = 64 KB

__global__ __launch_bounds__(256)
void roi_pool_kernel(const float* __restrict__ rois,
                     const float* __restrict__ conv,
                     float* __restrict__ out) {
    __shared__ float lds[CPB][H_][W_];        // 64 KB static LDS

    const int nCChunks = C_ / CPB;            // 64
    const int blk    = blockIdx.x;
    const int cchunk = blk % nCChunks;
    const int n      = (blk / nCChunks) % N_;
    const int b      = blk / (nCChunks * N_);
    const int c0     = cchunk * CPB;
    const int t      = threadIdx.x;

    // ---- ROI metadata (matches reference: trunc, expand, kernel/stride) ----
    const float* r4 = rois + (b * N_ + n) * 4;
    int x1 = (int)(r4[0] * (1.0f / ANCHOR_STRIDE));
    int y1 = (int)(r4[1] * (1.0f / ANCHOR_STRIDE));
    int x2 = (int)(r4[2] * (1.0f / ANCHOR_STRIDE));
    int y2 = (int)(r4[3] * (1.0f / ANCHOR_STRIDE));
#pragma unroll
    for (int it = 0; it < POOL; ++it) {
        if (y2 - y1 + 1 < POOL) { y1 = max(0, y1 - 1); y2 = min(H_ - 1, y2 + 1); }
    }
#pragma unroll
    for (int it = 0; it < POOL; ++it) {
        if (x2 - x1 + 1 < POOL) { x1 = max(0, x1 - 1); x2 = min(W_ - 1, x2 + 1); }
    }
    const int Lh = y2 - y1 + 1;               // 7..64
    const int Lw = x2 - x1 + 1;               // 7..64
    const int kh = (Lh + POOL - 1) / POOL;    // row kernel
    const int sh = Lh / POOL;                 // row stride
    const int kw = (Lw + POOL - 1) / POOL;    // col kernel
    const int sw = Lw / POOL;                 // col stride

    // ---- async-stage rows [y1..y2], full width, CPB channels, into LDS ----
    const int chunksPerRow  = (W_ * 4) / 16;          // 16 x 16B per row
    const int chunksPerChan = Lh * chunksPerRow;
    const int totalChunks   = CPB * chunksPerChan;    // <= 4096
    const int nIter         = (totalChunks + 255) / 256;

    const uint32_t ldsBase = (uint32_t)(uintptr_t)(&lds[0][0][0]);

    for (int k = 0; k < nIter; ++k) {
        int i = t + k * 256;
        if (i >= totalChunks) i = totalChunks - 1;    // duplicate xfer: harmless
        const int c_sub = i / chunksPerChan;
        int rem = i - c_sub * chunksPerChan;
        const int row = rem / chunksPerRow;           // relative to y1
        const int seg = rem - row * chunksPerRow;

        const uint32_t gOff =
            (uint32_t)((((b * C_ + c0 + c_sub) * H_ + (y1 + row)) * W_) * 4
                       + seg * 16);
        const uint32_t lOff =
            ldsBase + (uint32_t)(((c_sub * H_ + row) * W_) * 4 + seg * 16);

        // CDNA5: DMA 16B/lane global -> LDS, tracked by ASYNCcnt.
        asm volatile("global_load_async_to_lds_b128 %0, %1, %2"
                     :
                     : "v"(lOff), "v"(gOff), "s"(conv)
                     : "memory");
    }

    // Drain this wave's async transfers, then make them WG-visible.
    asm volatile("s_wait_asynccnt 0" ::: "memory");
    __syncthreads();

    // ---- pooled max from LDS: 4 channels x 7 x 7 outputs ----
    if (t < CPB * POOL * POOL) {
        const int c_sub = t / (POOL * POOL);
        const int rem   = t % (POOL * POOL);
        const int ph    = rem / POOL;
        const int pw    = rem % POOL;

        const int r0   = ph * sh;          // row index relative to y1
        const int col0 = x1 + pw * sw;     // absolute column (rows staged full-width)

        float m = -INFINITY;
        for (int r = 0; r < kh; ++r) {
            for (int s = 0; s < kw; ++s) {
                m = fmaxf(m, lds[c_sub][r0 + r][col0 + s]);
            }
        }
        out[(((b * N_ + n) * C_ + c0 + c_sub) * POOL + ph) * POOL + pw] = m;
    }
}

extern "C" void kernel_launch(void* const* d_in, const int* in_sizes, int n_in,
                              void* d_out, int out_size, void* d_ws, size_t ws_size,
                              hipStream_t stream) {
    (void)in_sizes; (void)n_in; (void)out_size; (void)d_ws; (void)ws_size;
    const float* rois = (const float*)d_in[0];   // [B,N,4]
    const float* conv = (const float*)d_in[1];   // [B,C,H,W]
    float* out = (float*)d_out;                  // [B,N,C,7,7]

    const int blocks = B_ * N_ * (C_ / CPB);     // 8192 workgroups
    roi_pool_kernel<<<blocks, 256, 0, stream>>>(rois, conv, out);
}